// RandomShift_35270271434805
// MI455X (gfx1250) — compile-verified
//
#include <hip/hip_runtime.h>

// RandomShift: out[b,t] = in[b, t-s_b] if 0 <= t-s_b < T else 0
// B=2048, T=16000, fp32. Pure streaming op: 262 MB traffic, ~11.3 us floor
// at 23.3 TB/s. Strategy: shift is uniform per row -> contiguous shifted
// memcpy per row with b128 vector accesses and non-temporal cache hints.

#define RS_BATCH 2048
#define RS_TIME  16000
#define RS_TPB   256

// 4-float vector with *4-byte* alignment: source addresses are shifted by an
// arbitrary element count, so they are dword-aligned but not always 16B
// aligned. gfx1250 supports unaligned (dword-aligned) b128 global access.
typedef float v4f __attribute__((ext_vector_type(4), aligned(4)));

__global__ __launch_bounds__(RS_TPB) void RandomShift_kernel(
    const float* __restrict__ in,
    const int*   __restrict__ shifts,
    float*       __restrict__ out)
{
    const int b = blockIdx.x;
    // Wave-uniform: lowers to a scalar load (s_load_b32).
    const int s = shifts[b];

    const float* __restrict__ rowIn  = in  + (size_t)b * RS_TIME;
    float*       __restrict__ rowOut = out + (size_t)b * RS_TIME;

    const int nvec = RS_TIME / 4;  // 4000 float4 chunks per row

    #pragma unroll 4
    for (int v = threadIdx.x; v < nvec; v += RS_TPB) {
        const int t0   = v * 4;       // output element index (16B aligned)
        const int src0 = t0 - s;      // source element index (4B aligned)

        v4f r;
        if (src0 >= 0 && src0 + 3 < RS_TIME) {
            // Common case: whole chunk in range -> one b128 NT load.
            r = __builtin_nontemporal_load(
                    reinterpret_cast<const v4f*>(rowIn + src0));
        } else {
            // Boundary / fully-padded chunks (at most 2 boundary chunks/row).
            #pragma unroll
            for (int k = 0; k < 4; ++k) {
                const int sk = src0 + k;
                r[k] = (sk >= 0 && sk < RS_TIME) ? rowIn[sk] : 0.0f;
            }
        }

        // Output chunk is 16B aligned; NT store -> streams past L2 (TH=NT).
        __builtin_nontemporal_store(r, reinterpret_cast<v4f*>(rowOut + t0));
    }
}

extern "C" void kernel_launch(void* const* d_in, const int* in_sizes, int n_in,
                              void* d_out, int out_size, void* d_ws, size_t ws_size,
                              hipStream_t stream)
{
    const float* inputs = (const float*)d_in[0];   // [2048, 16000] fp32
    const int*   shifts = (const int*)d_in[1];     // [2048] int32
    float*       out    = (float*)d_out;           // [2048, 16000] fp32

    (void)in_sizes; (void)n_in; (void)out_size; (void)d_ws; (void)ws_size;

    RandomShift_kernel<<<RS_BATCH, RS_TPB, 0, stream>>>(inputs, shifts, out);
}